// MultiBoxLoss_70033736729062
// MI455X (gfx1250) — compile-verified
//
#include <hip/hip_runtime.h>
#include <hip/hip_bf16.h>

// Problem constants (from reference setup_inputs)
#define BB 32
#define AA 32768
#define CC 81
#define GG 64

typedef __attribute__((ext_vector_type(2))) float v2f;
typedef __attribute__((ext_vector_type(8))) float v8f;

// ---------------------------------------------------------------------------
// Kernel 0: zero the scratch state that is accumulated into (packed argmax
// cells and per-batch positive counts). Everything else is fully overwritten.
// ---------------------------------------------------------------------------
__global__ void init_kernel(unsigned long long* packed, int* numPos) {
    int i = blockIdx.x * blockDim.x + threadIdx.x;
    if (i < BB * GG) packed[i] = 0ULL;
    if (i < BB) numPos[i] = 0;
}

// ---------------------------------------------------------------------------
// Kernel 1: IoU matching.
//  - per-anchor best gt (first-max tie rule, like jnp.argmax axis=1)
//  - per-gt best anchor via packed u64 max: (iou_bits<<32)|(~anchor_idx)
//    -> max picks highest IoU, lowest anchor index on ties (argmax axis=0).
// ---------------------------------------------------------------------------
__global__ void match_kernel(const float* __restrict__ anchors,
                             const float* __restrict__ gt_boxes,
                             float* __restrict__ bestIou,
                             int* __restrict__ bestIdx,
                             unsigned long long* __restrict__ packed) {
    const int blocksPerBatch = AA / 256;
    int b = blockIdx.x / blocksPerBatch;
    int a = (blockIdx.x % blocksPerBatch) * 256 + threadIdx.x;

    __shared__ float gtb[GG * 4];
    __shared__ unsigned long long lbest[GG];
    if (threadIdx.x < GG * 4) gtb[threadIdx.x] = gt_boxes[(size_t)b * GG * 4 + threadIdx.x];
    if (threadIdx.x < GG) lbest[threadIdx.x] = 0ULL;
    __syncthreads();

    float cx = anchors[a * 4 + 0], cy = anchors[a * 4 + 1];
    float w  = anchors[a * 4 + 2], h  = anchors[a * 4 + 3];
    float ax1 = cx - 0.5f * w, ay1 = cy - 0.5f * h;
    float ax2 = cx + 0.5f * w, ay2 = cy + 0.5f * h;
    float area1 = fmaxf(ax2 - ax1, 0.f) * fmaxf(ay2 - ay1, 0.f);

    float bi = -1.0f;
    int bg = 0;
    for (int g = 0; g < GG; ++g) {
        float gx1 = gtb[g * 4 + 0], gy1 = gtb[g * 4 + 1];
        float gx2 = gtb[g * 4 + 2], gy2 = gtb[g * 4 + 3];
        float area2 = fmaxf(gx2 - gx1, 0.f) * fmaxf(gy2 - gy1, 0.f);
        float lx = fmaxf(ax1, gx1), ly = fmaxf(ay1, gy1);
        float rx = fminf(ax2, gx2), ry = fminf(ay2, gy2);
        float iw = fmaxf(rx - lx, 0.f), ih = fmaxf(ry - ly, 0.f);
        float inter = iw * ih;
        float iou = inter / fmaxf(area1 + area2 - inter, 1e-6f);
        if (iou > bi) { bi = iou; bg = g; }
        unsigned long long pv =
            ((unsigned long long)__float_as_uint(iou) << 32) |
            (unsigned long long)(0xFFFFFFFFu - (unsigned)a);
        atomicMax(&lbest[g], pv);
    }
    int row = b * AA + a;
    bestIou[row] = bi;
    bestIdx[row] = bg;
    __syncthreads();
    if (threadIdx.x < GG) atomicMax(&packed[b * GG + threadIdx.x], lbest[threadIdx.x]);
}

// ---------------------------------------------------------------------------
// Kernel 2: force each gt's best anchor to match that gt (serial ascending g,
// last-write-wins, mirroring the torch loop). One block per batch.
// ---------------------------------------------------------------------------
__global__ void force_kernel(const unsigned long long* __restrict__ packed,
                             float* __restrict__ bestIou,
                             int* __restrict__ bestIdx) {
    int b = blockIdx.x;
    __shared__ int aidx[GG];
    if (threadIdx.x < GG) {
        unsigned long long p = packed[b * GG + threadIdx.x];
        aidx[threadIdx.x] = (int)(0xFFFFFFFFu - (unsigned)(p & 0xFFFFFFFFull));
    }
    __syncthreads();
    if (threadIdx.x == 0) {
        for (int g = 0; g < GG; ++g) {
            int ai = aidx[g];
            bestIdx[b * AA + ai] = g;
            bestIou[b * AA + ai] = 1.0f;
        }
    }
}

// ---------------------------------------------------------------------------
// Kernel 3: matched labels, box encoding, smooth-L1 loc loss (block partials),
// per-batch positive counts (integer atomics -> deterministic).
// ---------------------------------------------------------------------------
__global__ void loc_kernel(const float* __restrict__ pred_locs,
                           const float* __restrict__ anchors,
                           const float* __restrict__ gt_boxes,
                           const int* __restrict__ gt_labels,
                           const float* __restrict__ bestIou,
                           const int* __restrict__ bestIdx,
                           int* __restrict__ label,
                           int* __restrict__ numPos,
                           float* __restrict__ locPart) {
    int row = blockIdx.x * 256 + threadIdx.x;
    int b = row >> 15;              // AA = 2^15
    int a = row & (AA - 1);

    float iou = bestIou[row];
    int gi = bestIdx[row];
    int lab = (iou < 0.5f) ? 0 : gt_labels[b * GG + gi];
    label[row] = lab;
    int pos = lab > 0;

    float contrib = 0.0f;
    if (pos) {
        const float* gb = &gt_boxes[((size_t)b * GG + gi) * 4];
        float gx1 = gb[0], gy1 = gb[1], gx2 = gb[2], gy2 = gb[3];
        float gcx = 0.5f * (gx1 + gx2), gcy = 0.5f * (gy1 + gy2);
        float gw = gx2 - gx1, gh = gy2 - gy1;
        float acx = anchors[a * 4 + 0], acy = anchors[a * 4 + 1];
        float aw  = anchors[a * 4 + 2], ah  = anchors[a * 4 + 3];
        float t0 = (gcx - acx) / (0.1f * aw);
        float t1 = (gcy - acy) / (0.1f * ah);
        float t2 = logf(gw / fmaxf(aw, 1e-6f)) / 0.2f;
        float t3 = logf(gh / fmaxf(ah, 1e-6f)) / 0.2f;
        float tt[4] = {t0, t1, t2, t3};
        const float* pl = &pred_locs[(size_t)row * 4];
        for (int j = 0; j < 4; ++j) {
            float d = pl[j] - tt[j];
            float ad = fabsf(d);
            contrib += (ad < 1.0f) ? 0.5f * d * d : ad - 0.5f;
        }
    }

    __shared__ float red[256];
    __shared__ int cnt;
    if (threadIdx.x == 0) cnt = 0;
    red[threadIdx.x] = contrib;
    __syncthreads();
    if (pos) atomicAdd(&cnt, 1);
    for (int off = 128; off; off >>= 1) {
        if (threadIdx.x < off) red[threadIdx.x] += red[threadIdx.x + off];
        __syncthreads();
    }
    if (threadIdx.x == 0) {
        locPart[blockIdx.x] = red[0];
        atomicAdd(&numPos[b], cnt);
    }
}

// ---------------------------------------------------------------------------
// Kernel 4: the 340 MB streaming pass. One anchor per wave; lanes cooperate
// across C=81 (lane reads c = l, l+32, l+64) -> fully coalesced. Wave shuffle
// reductions for max / sum-exp / gather-at-label.
// ---------------------------------------------------------------------------
__global__ void conf_kernel(const float* __restrict__ pred_confs,
                            const int* __restrict__ label,
                            float* __restrict__ confNeg,
                            float* __restrict__ confPosPart) {
    int wave = threadIdx.x >> 5;
    int lane = threadIdx.x & 31;
    int row = blockIdx.x * 8 + wave;

    const float* x = &pred_confs[(size_t)row * CC];
    float v0 = x[lane];
    float v1 = x[lane + 32];
    bool has2 = lane < (CC - 64);   // lane < 17
    float v2 = has2 ? x[lane + 64] : -3.402823466e38f;

    float m = fmaxf(fmaxf(v0, v1), v2);
    for (int o = 16; o; o >>= 1) m = fmaxf(m, __shfl_xor(m, o, 32));

    int lab = label[row];
    float s = expf(v0 - m) + expf(v1 - m) + (has2 ? expf(v2 - m) : 0.0f);
    float xl = (lane == lab ? v0 : 0.0f);
    if (lane + 32 == lab) xl += v1;
    if (has2 && lane + 64 == lab) xl += v2;
    for (int o = 16; o; o >>= 1) {
        s  += __shfl_xor(s, o, 32);
        xl += __shfl_xor(xl, o, 32);
    }

    __shared__ float wp[8];
    if (lane == 0) {
        float lse = m + logf(s);
        float conf = lse - xl;
        int pos = lab > 0;
        confNeg[row] = pos ? 0.0f : conf;
        wp[wave] = pos ? conf : 0.0f;
    }
    __syncthreads();
    if (threadIdx.x == 0) {
        float t = 0.0f;
        for (int i = 0; i < 8; ++i) t += wp[i];
        confPosPart[blockIdx.x] = t;
    }
}

// ---------------------------------------------------------------------------
// Kernel 5: exact top-k sum of conf_loss_neg per batch via 4-pass radix
// select on the (monotone for >=0) float bit pattern. One block per batch.
// Replaces the reference's double argsort rank mask exactly (mod exact ties).
// ---------------------------------------------------------------------------
__global__ void topk_kernel(const float* __restrict__ confNeg,
                            const int* __restrict__ numPos,
                            float* __restrict__ topkPart) {
    int b = blockIdx.x;
    int tid = threadIdx.x;
    __shared__ unsigned hist[256];
    __shared__ unsigned sSel, sAbove;
    __shared__ float red[256];

    int np = numPos[b];
    long long k64 = 3LL * np;
    int k = (int)((k64 < (AA - 1)) ? k64 : (AA - 1));
    if (k <= 0) {
        if (tid == 0) topkPart[b] = 0.0f;
        return;
    }

    const float* v = &confNeg[(size_t)b * AA];
    unsigned prefix = 0;
    int remaining = k;
    for (int pass = 0; pass < 4; ++pass) {
        int shift = 24 - 8 * pass;
        unsigned hiMask = (pass == 0) ? 0u : (0xFFFFFFFFu << (32 - 8 * pass));
        hist[tid] = 0;
        __syncthreads();
        for (int i = tid; i < AA; i += 256) {
            unsigned u = __float_as_uint(v[i]);
            if ((u & hiMask) == prefix) atomicAdd(&hist[(u >> shift) & 255u], 1u);
        }
        __syncthreads();
        if (tid == 0) {
            unsigned cum = 0;
            int sel = 0;
            for (int bin = 255; bin >= 0; --bin) {
                if (cum + hist[bin] >= (unsigned)remaining) { sel = bin; break; }
                cum += hist[bin];
            }
            sSel = (unsigned)sel;
            sAbove = cum;
        }
        __syncthreads();
        prefix |= (sSel << shift);
        remaining -= (int)sAbove;
        __syncthreads();
    }

    // prefix = bit pattern of k-th largest value t.
    // topk sum = sum(v > t) + remaining * t   (remaining elements equal t)
    float t = __uint_as_float(prefix);
    float s = 0.0f;
    for (int i = tid; i < AA; i += 256) {
        float val = v[i];
        if (__float_as_uint(val) > prefix) s += val;
    }
    red[tid] = s;
    __syncthreads();
    for (int off = 128; off; off >>= 1) {
        if (tid < off) red[tid] += red[tid + off];
        __syncthreads();
    }
    if (tid == 0) topkPart[b] = red[0] + (float)remaining * t;
}

// ---------------------------------------------------------------------------
// Kernel 6: deterministic final reductions. 256->64 in LDS, then 64->1 with
// V_WMMA_F32_16X16X4_F32: A = values (16x4), B = ones (4x16)  => D rows are
// row-sums (exact f32 FMA). Sum the 8 accumulator VGPRs + shfl_xor(16).
// ---------------------------------------------------------------------------
__device__ __forceinline__ float wmma_sum64(const float* m64, int lane) {
    v2f a, bm;
    v8f c = {0.f, 0.f, 0.f, 0.f, 0.f, 0.f, 0.f, 0.f};
    int m = lane & 15;
    int k0 = (lane < 16) ? 0 : 2;            // A 16x4 layout: lanes 0-15 K=0,1; 16-31 K=2,3
    a[0] = m64[m * 4 + k0];
    a[1] = m64[m * 4 + k0 + 1];
    bm[0] = 1.0f;
    bm[1] = 1.0f;
    v8f d = __builtin_amdgcn_wmma_f32_16x16x4_f32(false, a, false, bm, (short)0, c, false, false);
    float p = d[0] + d[1] + d[2] + d[3] + d[4] + d[5] + d[6] + d[7]; // half the rows
    p += __shfl_xor(p, 16, 32);                                      // other half
    return p;
}

__device__ float grid_sum(const float* g, int n, float* red, float* m64) {
    int tid = threadIdx.x;
    float s = 0.0f;
    for (int i = tid; i < n; i += 256) s += g[i];   // fixed strided order
    red[tid] = s;
    __syncthreads();
    if (tid < 64) m64[tid] = red[tid] + red[tid + 64] + red[tid + 128] + red[tid + 192];
    __syncthreads();
    float r = 0.0f;
    if (tid < 32) r = wmma_sum64(m64, tid);         // wave 0 fully active: EXEC all 1s
    if (tid == 0) red[0] = r;
    __syncthreads();
    float out = red[0];
    __syncthreads();
    return out;
}

__global__ void final_kernel(const float* __restrict__ locPart, int nLoc,
                             const float* __restrict__ confPosPart, int nConf,
                             const float* __restrict__ topkPart,
                             const int* __restrict__ numPos,
                             float* __restrict__ out) {
    __shared__ float red[256];
    __shared__ float m64[64];
    float loc = grid_sum(locPart, nLoc, red, m64);
    float cp  = grid_sum(confPosPart, nConf, red, m64);
    if (threadIdx.x == 0) {
        float tk = 0.0f;
        int np = 0;
        for (int b = 0; b < BB; ++b) { tk += topkPart[b]; np += numPos[b]; }
        float N = (float)(np > 0 ? np : 1);
        float conf = cp + tk;
        out[0] = (loc + conf) / N;
        out[1] = loc / N;
        out[2] = conf / N;
    }
}

// ---------------------------------------------------------------------------
extern "C" void kernel_launch(void* const* d_in, const int* in_sizes, int n_in,
                              void* d_out, int out_size, void* d_ws, size_t ws_size,
                              hipStream_t stream) {
    const float* pred_locs  = (const float*)d_in[0];   // [B,A,4]
    const float* pred_confs = (const float*)d_in[1];   // [B,A,C]
    const float* anchors    = (const float*)d_in[2];   // [A,4]
    const float* gt_boxes   = (const float*)d_in[3];   // [B,G,4]
    const int*   gt_labels  = (const int*)  d_in[4];   // [B,G]
    float* out = (float*)d_out;

    // Workspace carve-up (256B aligned slices)
    char* base = (char*)d_ws;
    size_t off = 0;
    auto carve = [&](size_t bytes) -> char* {
        char* p = base + off;
        off = (off + bytes + 255) & ~(size_t)255;
        return p;
    };
    float* bestIou              = (float*)carve((size_t)BB * AA * 4);
    int*   bestIdx              = (int*)  carve((size_t)BB * AA * 4);
    int*   label                = (int*)  carve((size_t)BB * AA * 4);
    float* confNeg              = (float*)carve((size_t)BB * AA * 4);
    unsigned long long* packed  = (unsigned long long*)carve((size_t)BB * GG * 8);
    int*   numPos               = (int*)  carve((size_t)BB * 4);
    const int nLocPart  = (BB * AA) / 256;   // 4096
    const int nConfPart = (BB * AA) / 8;     // 131072
    float* locPart              = (float*)carve((size_t)nLocPart * 4);
    float* confPosPart          = (float*)carve((size_t)nConfPart * 4);
    float* topkPart             = (float*)carve((size_t)BB * 4);

    init_kernel<<<(BB * GG + 255) / 256, 256, 0, stream>>>(packed, numPos);

    match_kernel<<<(BB * AA) / 256, 256, 0, stream>>>(
        anchors, gt_boxes, bestIou, bestIdx, packed);

    force_kernel<<<BB, 64, 0, stream>>>(packed, bestIou, bestIdx);

    loc_kernel<<<(BB * AA) / 256, 256, 0, stream>>>(
        pred_locs, anchors, gt_boxes, gt_labels, bestIou, bestIdx,
        label, numPos, locPart);

    conf_kernel<<<(BB * AA) / 8, 256, 0, stream>>>(
        pred_confs, label, confNeg, confPosPart);

    topk_kernel<<<BB, 256, 0, stream>>>(confNeg, numPos, topkPart);

    final_kernel<<<1, 256, 0, stream>>>(
        locPart, nLocPart, confPosPart, nConfPart, topkPart, numPos, out);
}